// CRF_16990890623007
// MI455X (gfx1250) — compile-verified
//
#include <hip/hip_runtime.h>

#define CRF_S     512
#define CRF_N     64
#define START_TAG 1
#define END_TAG   63
#define NEG_INF   (-10000.0f)
#define ROWSTRIDE ((size_t)CRF_S * CRF_N)   // feats elements per batch row

typedef float v2f __attribute__((ext_vector_type(2)));
typedef float v8f __attribute__((ext_vector_type(8)));

// LDS (per 16-batch block): raw T (gold/terminal gathers), fv/U padded to
// stride 68 so A-fragment loads and C-tile scatter are bank-conflict-free.
struct CrfShared {
  float T[64][64];     // raw transitions
  float fv[16][68];    // forward state (row-major, for phase 1/2 + terminal)
  float U[16][68];     // exp(fv - rowmax)
  float Mrow[16];
  float mcur[16];
  float gold[16];
};

__global__ __launch_bounds__(128) void crf_forward_kernel(
    const float* __restrict__ feats, const float* __restrict__ mask,
    const int* __restrict__ tags, const float* __restrict__ trans,
    float* __restrict__ ws)
{
  __shared__ CrfShared sm;
  const int tid = threadIdx.x;          // 128 threads = 4 waves (wave32)
  const int b0  = blockIdx.x * 16;      // 16 batches per block
  const int r   = tid >> 3;             // row 0..15 for phases 1/2
  const int q   = tid & 7;
  const int c0  = q * 8;

  // WMMA lane mapping (wave32 16x16x4 f32 fragments)
  const int lane = tid & 31;
  const int wv   = tid >> 5;            // wave -> N tile 0..3
  const int hw   = lane >> 4;           // half-wave
  const int am   = lane & 15;           // A row (M) / B,C column-in-tile
  const int ncol = (wv << 4) + am;      // global tag column

  // ---- init LDS: transitions + initial state ----
  for (int i = tid; i < 64 * 64; i += 128) sm.T[i >> 6][i & 63] = trans[i];
  for (int i = tid; i < 16 * 64; i += 128)
    sm.fv[i >> 6][i & 63] = ((i & 63) == START_TAG) ? 0.0f : NEG_INF;
  __syncthreads();

  // ---- hoist loop-invariant B fragments: E = exp(T), 16 chunks x v2f ----
  v2f bfrag[16];
  #pragma unroll
  for (int c = 0; c < 16; ++c) {
    const int k0 = 4 * c + (hw << 1);
    v2f b;
    b.x = __expf(sm.T[k0][ncol]);
    b.y = __expf(sm.T[k0 + 1][ncol]);
    bfrag[c] = b;
  }

  // ---- register-resident state (C-tile layout) ----
  float fvreg[8];
  #pragma unroll
  for (int vr = 0; vr < 8; ++vr)
    fvreg[vr] = (ncol == START_TAG) ? 0.0f : NEG_INF;

  // ---- emission pipeline: direct global loads in C-tile layout, 1 step ahead
  const float* eptr = feats + (size_t)(b0 + (hw << 3)) * ROWSTRIDE + ncol;
  float em[8];
  #pragma unroll
  for (int vr = 0; vr < 8; ++vr) em[vr] = eptr[(size_t)vr * ROWSTRIDE];

  // ---- gold-path pipeline (lanes 0..15 of wave 0) ----
  int   prevtag = START_TAG, tagreg = 0, cnt = 0;
  float gold = 0.0f, mreg = 0.0f;
  if (tid < 16) {
    tagreg = tags[(size_t)(b0 + tid) * CRF_S];
    mreg   = mask[(size_t)(b0 + tid) * CRF_S];
  }

  for (int s = 0; s < CRF_S; ++s) {
    // stage mask row; issue gold emission gather early (latency overlap)
    float ge = 0.0f;
    if (tid < 16) {
      sm.mcur[tid] = mreg;
      ge = feats[(size_t)(b0 + tid) * ROWSTRIDE + (size_t)s * CRF_N + tagreg];
    }

    // phase 1: per-batch row max (LDS reads + shfl butterfly over 8 lanes)
    float fl[8];
    {
      const float* fr = &sm.fv[r][c0];
      #pragma unroll
      for (int i = 0; i < 8; ++i) fl[i] = fr[i];
    }
    float mx = fl[0];
    #pragma unroll
    for (int i = 1; i < 8; ++i) mx = fmaxf(mx, fl[i]);
    mx = fmaxf(mx, __shfl_xor(mx, 1));
    mx = fmaxf(mx, __shfl_xor(mx, 2));
    mx = fmaxf(mx, __shfl_xor(mx, 4));
    if (q == 0) sm.Mrow[r] = mx;

    // phase 2: U = exp(fv - rowmax)
    {
      float* ur = &sm.U[r][c0];
      #pragma unroll
      for (int i = 0; i < 8; ++i) ur[i] = __expf(fl[i] - mx);
    }
    __syncthreads();

    // phase 3: V = U(16x64) x E(64x16) -- 16 WMMAs in 4 independent chains
    v8f acc0 = {}, acc1 = {}, acc2 = {}, acc3 = {};
    #pragma unroll
    for (int c = 0; c < 16; c += 4) {
      const int k0 = 4 * c + (hw << 1);
      v2f a0, a1, a2, a3;
      a0.x = sm.U[am][k0 + 0];  a0.y = sm.U[am][k0 + 1];
      a1.x = sm.U[am][k0 + 4];  a1.y = sm.U[am][k0 + 5];
      a2.x = sm.U[am][k0 + 8];  a2.y = sm.U[am][k0 + 9];
      a3.x = sm.U[am][k0 + 12]; a3.y = sm.U[am][k0 + 13];
      acc0 = __builtin_amdgcn_wmma_f32_16x16x4_f32(false, a0, false, bfrag[c + 0],
                                                   (short)0, acc0, false, false);
      acc1 = __builtin_amdgcn_wmma_f32_16x16x4_f32(false, a1, false, bfrag[c + 1],
                                                   (short)0, acc1, false, false);
      acc2 = __builtin_amdgcn_wmma_f32_16x16x4_f32(false, a2, false, bfrag[c + 2],
                                                   (short)0, acc2, false, false);
      acc3 = __builtin_amdgcn_wmma_f32_16x16x4_f32(false, a3, false, bfrag[c + 3],
                                                   (short)0, acc3, false, false);
    }
    const v8f acct = (acc0 + acc1) + (acc2 + acc3);

    // phase 4: fv = m*(log(V)+M+emit) + (1-m)*fv ; keep regs + mirror to LDS
    #pragma unroll
    for (int vr = 0; vr < 8; ++vr) {
      const int m = vr + (hw << 3);
      float nv = __logf(acct[vr]) + sm.Mrow[m] + em[vr];
      float mm = sm.mcur[m];
      fvreg[vr] = mm * nv + (1.0f - mm) * fvreg[vr];
      sm.fv[m][ncol] = fvreg[vr];
    }

    // gold accumulate + prefetch next step's tag/mask
    if (tid < 16) {
      gold += (ge + sm.T[prevtag][tagreg]) * mreg;
      cnt  += (mreg != 0.0f) ? 1 : 0;
      prevtag = tagreg;
      if (s + 1 < CRF_S) {
        tagreg = tags[(size_t)(b0 + tid) * CRF_S + s + 1];
        mreg   = mask[(size_t)(b0 + tid) * CRF_S + s + 1];
      }
    }

    // prefetch next step's emissions into registers
    if (s + 1 < CRF_S) {
      eptr += CRF_N;
      #pragma unroll
      for (int vr = 0; vr < 8; ++vr) em[vr] = eptr[(size_t)vr * ROWSTRIDE];
      if (s + 2 < CRF_S) __builtin_prefetch(eptr + CRF_N, 0, 0);
    }
    __syncthreads();
  }

  // ---- terminal: forward = logsumexp(fv + T[:, END]) per batch ----
  float tv[8];
  float tmx = -3.0e38f;
  {
    const float* fr = &sm.fv[r][c0];
    #pragma unroll
    for (int i = 0; i < 8; ++i) {
      tv[i] = fr[i] + sm.T[c0 + i][END_TAG];
      tmx = fmaxf(tmx, tv[i]);
    }
    tmx = fmaxf(tmx, __shfl_xor(tmx, 1));
    tmx = fmaxf(tmx, __shfl_xor(tmx, 2));
    tmx = fmaxf(tmx, __shfl_xor(tmx, 4));
  }
  float ssum = 0.0f;
  #pragma unroll
  for (int i = 0; i < 8; ++i) ssum += __expf(tv[i] - tmx);
  ssum += __shfl_xor(ssum, 1);
  ssum += __shfl_xor(ssum, 2);
  ssum += __shfl_xor(ssum, 4);
  const float fwd = __logf(ssum) + tmx;

  if (tid < 16) {
    int idx  = cnt - 1;
    int last = (idx >= 0) ? tags[(size_t)(b0 + tid) * CRF_S + idx] : START_TAG;
    sm.gold[tid] = gold + sm.T[last][END_TAG];
  }
  __syncthreads();
  if (q == 0) ws[b0 + r] = fwd - sm.gold[r];
}

// deterministic mean reduction (no float atomics)
__global__ __launch_bounds__(256) void crf_mean_kernel(
    const float* __restrict__ ws, float* __restrict__ out, int n)
{
  __shared__ float buf[256];
  float s = 0.0f;
  for (int i = threadIdx.x; i < n; i += 256) s += ws[i];
  buf[threadIdx.x] = s;
  __syncthreads();
  for (int off = 128; off > 0; off >>= 1) {
    if (threadIdx.x < off) buf[threadIdx.x] += buf[threadIdx.x + off];
    __syncthreads();
  }
  if (threadIdx.x == 0) out[0] = buf[0] / (float)n;
}

extern "C" void kernel_launch(void* const* d_in, const int* in_sizes, int n_in,
                              void* d_out, int out_size, void* d_ws, size_t ws_size,
                              hipStream_t stream) {
  const float* feats = (const float*)d_in[0];
  const float* mask  = (const float*)d_in[1];
  const int*   tags  = (const int*)d_in[2];
  const float* trans = (const float*)d_in[3];

  const int BS = in_sizes[1];           // B * S
  const int B  = BS / CRF_S;            // S fixed at 512 per reference
  float* ws = (float*)d_ws;             // per-batch (forward - gold)

  crf_forward_kernel<<<B / 16, 128, 0, stream>>>(feats, mask, tags, trans, ws);
  crf_mean_kernel<<<1, 256, 0, stream>>>(ws, (float*)d_out, B);
}